// CRF_BiLSTM_69320772158150
// MI455X (gfx1250) — compile-verified
//
#include <hip/hip_runtime.h>

typedef __attribute__((ext_vector_type(2))) float v2f;
typedef __attribute__((ext_vector_type(8))) float v8f;

#define S_LEN 128      // scan length (original B)
#define N_BATCH 512    // lstm "batch" (original T)
#define DIN 768
#define H 128
#define G4 512         // 4*H gate columns
#define ROWS 65536     // S_LEN*N_BATCH
#define K_TAGS 13
#define W_STRIDE 516   // padded LDS stride for transposed w_hh [k][g]
#define H_STRIDE 132   // padded LDS stride for h [n][k]

// ---------------------------------------------------------------------------
// Kernel 1: pre[row][g] = sum_d X[row][d]*W[g][d] + bih[g] + bhh[g]
// M=65536, N=512, K=768. Block = 256 thr (8 waves), tile 64x128, K-step 16.
// blockIdx.z selects direction (weight/bias/output set).
// Register double-buffering: global loads for K-tile kt+1 overlap WMMAs of kt.
// ---------------------------------------------------------------------------
__global__ __launch_bounds__(256) void bilstm_gemm_pre(
    const float* __restrict__ X,
    const float* __restrict__ W0, const float* __restrict__ bi0,
    const float* __restrict__ bh0, float* __restrict__ pre0,
    const float* __restrict__ W1, const float* __restrict__ bi1,
    const float* __restrict__ bh1, float* __restrict__ pre1)
{
  const int z = blockIdx.z;
  const float* __restrict__ W   = z ? W1 : W0;
  const float* __restrict__ bih = z ? bi1 : bi0;
  const float* __restrict__ bhh = z ? bh1 : bh0;
  float* __restrict__ pre       = z ? pre1 : pre0;

  __shared__ float lds_a[64 * 17];
  __shared__ float lds_b[16 * 129];
  const int tid  = threadIdx.x;
  const int lane = tid & 31;
  const int hi   = lane >> 4;      // 0: lanes 0-15, 1: lanes 16-31
  const int l15  = lane & 15;
  const int wave = tid >> 5;
  const int wm   = wave & 3;       // 4 row strips of 16
  const int wn   = wave >> 2;      // 2 col strips of 64
  const int m0   = blockIdx.y * 64;
  const int n0   = blockIdx.x * 128;

  v8f acc[4];
  v8f zero = {};
  #pragma unroll
  for (int t = 0; t < 4; ++t) acc[t] = zero;

  const int la_r = tid >> 2;          // 0..63
  const int la_c = (tid & 3) * 4;     // 0,4,8,12
  const int lb_k = tid & 15;
  const int lb_g = tid >> 4;          // 0..15

  // prologue: load K-tile 0 into registers
  float4 ra;
  float  rb[8];
  {
    ra = *(const float4*)(X + (size_t)(m0 + la_r) * DIN + la_c);
    #pragma unroll
    for (int p = 0; p < 8; ++p)
      rb[p] = W[(size_t)(n0 + p * 16 + lb_g) * DIN + lb_k];
  }

  for (int kt = 0; kt < 48; ++kt) {
    // commit current registers to LDS
    lds_a[la_r * 17 + la_c + 0] = ra.x;
    lds_a[la_r * 17 + la_c + 1] = ra.y;
    lds_a[la_r * 17 + la_c + 2] = ra.z;
    lds_a[la_r * 17 + la_c + 3] = ra.w;
    #pragma unroll
    for (int p = 0; p < 8; ++p)
      lds_b[lb_k * 129 + p * 16 + lb_g] = rb[p];
    __syncthreads();

    // prefetch next K-tile into registers (clamped; overlap with WMMA below)
    const int ktn = (kt + 1 < 48) ? kt + 1 : 47;
    const int k0n = ktn * 16;
    ra = *(const float4*)(X + (size_t)(m0 + la_r) * DIN + k0n + la_c);
    #pragma unroll
    for (int p = 0; p < 8; ++p)
      rb[p] = W[(size_t)(n0 + p * 16 + lb_g) * DIN + k0n + lb_k];

    #pragma unroll
    for (int kk = 0; kk < 16; kk += 4) {
      const int ak = kk + hi * 2;        // K split across lane halves
      v2f a;
      a.x = lds_a[(wm * 16 + l15) * 17 + ak];
      a.y = lds_a[(wm * 16 + l15) * 17 + ak + 1];
      #pragma unroll
      for (int t = 0; t < 4; ++t) {
        const int gl = wn * 64 + t * 16 + l15;
        v2f b;
        b.x = lds_b[ak * 129 + gl];
        b.y = lds_b[(ak + 1) * 129 + gl];
        acc[t] = __builtin_amdgcn_wmma_f32_16x16x4_f32(
            false, a, false, b, (short)0, acc[t], false, false);
      }
    }
    __syncthreads();
  }
  // D layout: VGPR r -> row r (lanes 0-15) / row r+8 (lanes 16-31)
  #pragma unroll
  for (int t = 0; t < 4; ++t) {
    const int gc = n0 + wn * 64 + t * 16 + l15;
    const float bias = bih[gc] + bhh[gc];
    const int rbase = m0 + wm * 16 + hi * 8;
    #pragma unroll
    for (int r = 0; r < 8; ++r)
      pre[(size_t)(rbase + r) * G4 + gc] = acc[t][r] + bias;
  }
}

// ---------------------------------------------------------------------------
// Kernel 2: LSTM recurrence. Block = 16 independent sequences, 128 steps.
// Per step: gates(16x512) = pre + h(16x128) @ w_hh^T via WMMA, then i/f/g/o.
// dir_force < 0: blockIdx.y selects direction (both run concurrently).
// Next step's pre tile prefetched into registers during the WMMA phase.
// Stores h trajectory of n==511 into `last` (fwd cols 0..127, bwd 128..255).
// ---------------------------------------------------------------------------
__global__ __launch_bounds__(256) void bilstm_rec(
    const float* __restrict__ pre_f, const float* __restrict__ whh_f,
    const float* __restrict__ pre_b, const float* __restrict__ whh_b,
    float* __restrict__ last, int dir_force)
{
  extern __shared__ float sm[];
  float* w_lds = sm;                        // 128*516 floats (w^T, [k][g])
  float* g_lds = sm + 128 * W_STRIDE;       // 16*512 floats (pre -> gates)
  float* h_lds = g_lds + 16 * G4;           // 16*132 floats
  const int dir = (dir_force < 0) ? (int)blockIdx.y : dir_force;
  const int reverse = dir;
  const float* __restrict__ pre = dir ? pre_b : pre_f;
  const float* __restrict__ whh = dir ? whh_b : whh_f;

  const int tid  = threadIdx.x;
  const int lane = tid & 31;
  const int hi   = lane >> 4;
  const int l15  = lane & 15;
  const int wave = tid >> 5;
  const int n0   = blockIdx.x * 16;

  for (int i = tid; i < G4 * H; i += 256) {
    const int g = i >> 7, k = i & 127;
    w_lds[k * W_STRIDE + g] = whh[i];       // whh[g][k] -> wt[k][g]
  }
  for (int i = tid; i < 16 * H_STRIDE; i += 256) h_lds[i] = 0.f;
  float c_reg[8];
  #pragma unroll
  for (int r = 0; r < 8; ++r) c_reg[r] = 0.f;
  const int e_n  = tid >> 4;                // elementwise: n row 0..15
  const int e_h0 = (tid & 15) * 8;          // 8 consecutive h-dims
  const int g_base = wave * 64;             // wave's 64 gate cols
  const bool is_last = (n0 + e_n) == (N_BATCH - 1);

  // prologue: fetch first pre tile (16 rows contiguous: 2048 float4)
  float4 pf[8];
  {
    const int s0 = reverse ? (S_LEN - 1) : 0;
    const float4* src = (const float4*)(pre + ((size_t)s0 * N_BATCH + n0) * G4);
    #pragma unroll
    for (int j = 0; j < 8; ++j) pf[j] = src[tid + j * 256];
  }
  __syncthreads();   // covers w_lds/h_lds init
  {
    float4* dst = (float4*)g_lds;
    #pragma unroll
    for (int j = 0; j < 8; ++j) dst[tid + j * 256] = pf[j];
  }
  __syncthreads();

  for (int it = 0; it < S_LEN; ++it) {
    const int s = reverse ? (S_LEN - 1 - it) : it;
    // issue prefetch of next step's tile (clamped on last iteration)
    {
      const int itn = (it + 1 < S_LEN) ? it + 1 : it;
      const int sn = reverse ? (S_LEN - 1 - itn) : itn;
      const float4* src = (const float4*)(pre + ((size_t)sn * N_BATCH + n0) * G4);
      #pragma unroll
      for (int j = 0; j < 8; ++j) pf[j] = src[tid + j * 256];
    }

    v8f acc[4];
    #pragma unroll
    for (int t = 0; t < 4; ++t) {           // C init = pre in D layout
      const int gc = g_base + t * 16 + l15;
      v8f a;
      #pragma unroll
      for (int r = 0; r < 8; ++r) a[r] = g_lds[(hi * 8 + r) * G4 + gc];
      acc[t] = a;
    }
    for (int k = 0; k < H; k += 4) {
      const int ak = k + hi * 2;
      v2f av;
      av.x = h_lds[l15 * H_STRIDE + ak];
      av.y = h_lds[l15 * H_STRIDE + ak + 1];
      #pragma unroll
      for (int t = 0; t < 4; ++t) {
        const int gc = g_base + t * 16 + l15;
        v2f bv;
        bv.x = w_lds[ak * W_STRIDE + gc];
        bv.y = w_lds[(ak + 1) * W_STRIDE + gc];
        acc[t] = __builtin_amdgcn_wmma_f32_16x16x4_f32(
            false, av, false, bv, (short)0, acc[t], false, false);
      }
    }
    #pragma unroll
    for (int t = 0; t < 4; ++t) {           // write gates back (own region)
      const int gc = g_base + t * 16 + l15;
      #pragma unroll
      for (int r = 0; r < 8; ++r) g_lds[(hi * 8 + r) * G4 + gc] = acc[t][r];
    }
    __syncthreads();

    #pragma unroll
    for (int r = 0; r < 8; ++r) {
      const int hk = e_h0 + r;
      float iv = g_lds[e_n * G4 + hk];
      float fv = g_lds[e_n * G4 + H + hk];
      float gv = g_lds[e_n * G4 + 2 * H + hk];
      float ov = g_lds[e_n * G4 + 3 * H + hk];
      iv = 1.f / (1.f + expf(-iv));
      fv = 1.f / (1.f + expf(-fv));
      ov = 1.f / (1.f + expf(-ov));
      const float cc = fv * c_reg[r] + iv * tanhf(gv);
      c_reg[r] = cc;
      const float hh = ov * tanhf(cc);
      h_lds[e_n * H_STRIDE + hk] = hh;
      if (is_last) last[s * 256 + (reverse ? H : 0) + hk] = hh;
    }
    __syncthreads();

    // commit prefetched tile for next step
    {
      float4* dst = (float4*)g_lds;
      #pragma unroll
      for (int j = 0; j < 8; ++j) dst[tid + j * 256] = pf[j];
    }
    __syncthreads();
  }
}

// ---------------------------------------------------------------------------
// Kernel 3: MLP head + CRF NLL (tiny, one block). Deterministic reduction.
// ---------------------------------------------------------------------------
__global__ __launch_bounds__(256) void bilstm_mlp_crf(
    const float* __restrict__ last,
    const float* __restrict__ W1, const float* __restrict__ b1,
    const float* __restrict__ W2, const float* __restrict__ b2,
    const int* __restrict__ labels,
    const float* __restrict__ cstart, const float* __restrict__ cend,
    const float* __restrict__ ctrans,
    float* __restrict__ out)
{
  extern __shared__ float sm2[];
  float* hid = sm2;                 // 128*128
  float* em  = hid + 128 * 128;     // 128*16 (13 used)
  float* tr  = em + 128 * 16;       // 169 (+pad)
  float* red = tr + 176;            // 8
  const int tid  = threadIdx.x;
  const int lane = tid & 31;
  const int wave = tid >> 5;

  for (int idx = tid; idx < 128 * 128; idx += 256) {
    const int sR = idx >> 7, j = idx & 127;
    float a = b1[j];
    const float* lr = last + sR * 256;
    const float* wr = W1 + j * 256;
    for (int k = 0; k < 256; ++k) a += lr[k] * wr[k];
    hid[idx] = fmaxf(a, 0.f);
  }
  if (tid < K_TAGS * K_TAGS) tr[tid] = ctrans[tid];
  __syncthreads();
  for (int idx = tid; idx < 128 * K_TAGS; idx += 256) {
    const int sR = idx / K_TAGS, k2 = idx % K_TAGS;
    float a = b2[k2];
    const float* hr = hid + sR * 128;
    const float* wr = W2 + k2 * 128;
    for (int j = 0; j < 128; ++j) a += hr[j] * wr[j];
    em[sR * 16 + k2] = a;
  }
  __syncthreads();

  float local = 0.f;
  for (int bi = 0; bi < 16; ++bi) {      // each wave: 16 batch rows
    const int b = wave * 16 + bi;
    const float* e = em + b * 16;
    const int* lab = labels + b * 512;
    // gold path score (emissions + transitions), lane-strided then reduced
    float sc = 0.f;
    for (int t = lane; t < 512; t += 32) sc += e[lab[t]];
    for (int t = lane; t < 511; t += 32) sc += tr[lab[t] * K_TAGS + lab[t + 1]];
    for (int off = 16; off > 0; off >>= 1) sc += __shfl_down(sc, off, 32);
    // alpha recursion: lane k < 13 holds alpha_k
    float alpha = (lane < K_TAGS) ? (cstart[lane] + e[lane]) : -1e30f;
    for (int t = 1; t < 512; ++t) {
      float aj[K_TAGS];
      #pragma unroll
      for (int j = 0; j < K_TAGS; ++j) aj[j] = __shfl(alpha, j, 32);
      float na = -1e30f;
      if (lane < K_TAGS) {
        float m = -1e30f;
        #pragma unroll
        for (int j = 0; j < K_TAGS; ++j) m = fmaxf(m, aj[j] + tr[j * K_TAGS + lane]);
        float ss = 0.f;
        #pragma unroll
        for (int j = 0; j < K_TAGS; ++j) ss += expf(aj[j] + tr[j * K_TAGS + lane] - m);
        na = m + logf(ss) + e[lane];
      }
      alpha = na;
    }
    float v = (lane < K_TAGS) ? (alpha + cend[lane]) : -1e30f;
    float m = v;
    for (int off = 16; off > 0; off >>= 1) m = fmaxf(m, __shfl_xor(m, off, 32));
    float sx = expf(v - m);
    for (int off = 16; off > 0; off >>= 1) sx += __shfl_xor(sx, off, 32);
    const float logZ = m + logf(sx);
    if (lane == 0) {
      const float score = sc + cstart[lab[0]] + cend[lab[511]];
      local += logZ - score;             // accumulates -llh contribution
    }
  }
  if (lane == 0) red[wave] = local;
  __syncthreads();
  if (tid == 0) {
    float tot = 0.f;
    for (int w = 0; w < 8; ++w) tot += red[w];
    out[0] = tot;
  }
}

// ---------------------------------------------------------------------------
extern "C" void kernel_launch(void* const* d_in, const int* in_sizes, int n_in,
                              void* d_out, int out_size, void* d_ws, size_t ws_size,
                              hipStream_t stream) {
  const float* x      = (const float*)d_in[0];
  const int*   labels = (const int*)  d_in[1];
  const float* w_ih_f = (const float*)d_in[2];
  const float* w_hh_f = (const float*)d_in[3];
  const float* b_ih_f = (const float*)d_in[4];
  const float* b_hh_f = (const float*)d_in[5];
  const float* w_ih_b = (const float*)d_in[6];
  const float* w_hh_b = (const float*)d_in[7];
  const float* b_ih_b = (const float*)d_in[8];
  const float* b_hh_b = (const float*)d_in[9];
  const float* W1     = (const float*)d_in[10];
  const float* b1     = (const float*)d_in[11];
  const float* W2     = (const float*)d_in[12];
  const float* b2     = (const float*)d_in[13];
  const float* cstart = (const float*)d_in[14];
  const float* cend   = (const float*)d_in[15];
  const float* ctrans = (const float*)d_in[16];

  const size_t preElems  = (size_t)ROWS * G4;      // 33.5M floats per dir
  const size_t lastElems = 128 * 256;
  const size_t need2 = (2 * preElems + lastElems) * sizeof(float);
  const size_t need1 = (preElems + lastElems) * sizeof(float);

  const size_t smem_rec = (size_t)(128 * W_STRIDE + 16 * G4 + 16 * H_STRIDE) * sizeof(float);
  const size_t smem_crf = (size_t)(128 * 128 + 128 * 16 + 176 + 8) * sizeof(float);

  if (ws_size >= need2) {
    // both directions concurrent: 2 pre buffers, merged launches
    float* pre0  = (float*)d_ws;
    float* pre1  = pre0 + preElems;
    float* lastb = pre1 + preElems;
    dim3 ggrid(4, 1024, 2);
    bilstm_gemm_pre<<<ggrid, 256, 0, stream>>>(
        x, w_ih_f, b_ih_f, b_hh_f, pre0, w_ih_b, b_ih_b, b_hh_b, pre1);
    bilstm_rec<<<dim3(32, 2), 256, smem_rec, stream>>>(
        pre0, w_hh_f, pre1, w_hh_b, lastb, -1);
    bilstm_mlp_crf<<<1, 256, smem_crf, stream>>>(
        lastb, W1, b1, W2, b2, labels, cstart, cend, ctrans, (float*)d_out);
  } else if (ws_size >= need1) {
    // fallback: single pre buffer, directions serialized
    float* pre   = (float*)d_ws;
    float* lastb = pre + preElems;
    dim3 ggrid(4, 1024, 1);
    bilstm_gemm_pre<<<ggrid, 256, 0, stream>>>(
        x, w_ih_f, b_ih_f, b_hh_f, pre, w_ih_f, b_ih_f, b_hh_f, pre);
    bilstm_rec<<<dim3(32, 1), 256, smem_rec, stream>>>(
        pre, w_hh_f, pre, w_hh_f, lastb, 0);
    bilstm_gemm_pre<<<ggrid, 256, 0, stream>>>(
        x, w_ih_b, b_ih_b, b_hh_b, pre, w_ih_b, b_ih_b, b_hh_b, pre);
    bilstm_rec<<<dim3(32, 1), 256, smem_rec, stream>>>(
        pre, w_hh_b, pre, w_hh_b, lastb, 1);
    bilstm_mlp_crf<<<1, 256, smem_crf, stream>>>(
        lastb, W1, b1, W2, b2, labels, cstart, cend, ctrans, (float*)d_out);
  }
}